// VAE_57509612094057
// MI455X (gfx1250) — compile-verified
//
#include <hip/hip_runtime.h>
#include <math.h>

typedef __attribute__((ext_vector_type(2))) float v2f;
typedef __attribute__((ext_vector_type(8))) float v8f;

#define HIDN 100
#define WT2_LD 104
#define WT2_COLS 112

// sigmoid with native v_exp_f32 + v_rcp_f32 (no IEEE div expansion)
__device__ __forceinline__ float sigf(float a) {
  return __builtin_amdgcn_rcpf(1.0f + __expf(-a));
}

// ---------------------------------------------------------------------------
// Encoder: h2 = swish(swish(x@W1+b1)@W2+b2)  via V_WMMA_F32_16X16X4_F32
// One wave = one 16-sample tile. Layer-1 computed directly in WMMA A layout.
// ---------------------------------------------------------------------------
__global__ __launch_bounds__(256) void enc_wmma_kernel(
    const float* __restrict__ x,
    const float* __restrict__ w1, const float* __restrict__ b1,
    const float* __restrict__ w2, const float* __restrict__ b2,
    float* __restrict__ h2buf, int n)
{
  __shared__ float lw1[400];                 // [0,300) w1 (3x100), [300,400) b1
  __shared__ float lb2[HIDN];
  __shared__ float wt2[WT2_COLS * WT2_LD];   // wt2[col*104 + k] = w2[k][col], zero padded

  const int tid = threadIdx.x;
  for (int i = tid; i < WT2_COLS * WT2_LD; i += 256) wt2[i] = 0.0f;
  for (int i = tid; i < 300; i += 256) lw1[i] = w1[i];
  for (int i = tid; i < HIDN; i += 256) { lw1[300 + i] = b1[i]; lb2[i] = b2[i]; }
  __syncthreads();
  for (int i = tid; i < HIDN * HIDN; i += 256) {
    int k = i / HIDN;
    int c = i - k * HIDN;
    wt2[c * WT2_LD + k] = w2[i];
  }
  __syncthreads();

  const int wave = tid >> 5;
  const int lane = tid & 31;
  const int tile = blockIdx.x * 8 + wave;
  if (tile * 16 >= n) return;
  const int s    = lane & 15;
  const int koff = (lane < 16) ? 0 : 2;     // A/B layout: hi half-wave holds K+2,K+3
  const int sample = tile * 16 + s;
  float x0 = 0.f, x1 = 0.f, x2 = 0.f;
  if (sample < n) { x0 = x[sample*3+0]; x1 = x[sample*3+1]; x2 = x[sample*3+2]; }

  // layer 1, produced in the 16x4-per-chunk A-operand layout
  float hA[50];
#pragma unroll
  for (int c = 0; c < 25; ++c) {
#pragma unroll
    for (int j = 0; j < 2; ++j) {
      int k = 4*c + koff + j;
      float a = fmaf(x0, lw1[k], fmaf(x1, lw1[100+k], fmaf(x2, lw1[200+k], lw1[300+k])));
      hA[2*c+j] = a * sigf(a);
    }
  }

  v8f acc[7];
#pragma unroll
  for (int nt = 0; nt < 7; ++nt) acc[nt] = (v8f)(0.0f);

  const int col = lane & 15;
#pragma unroll
  for (int c = 0; c < 25; ++c) {
    v2f a;
    a.x = hA[2*c];
    a.y = hA[2*c+1];
#pragma unroll
    for (int nt = 0; nt < 7; ++nt) {
      v2f b = *(const v2f*)&wt2[(nt*16 + col) * WT2_LD + 4*c + koff];
      acc[nt] = __builtin_amdgcn_wmma_f32_16x16x4_f32(
          false, a, false, b, (short)0, acc[nt], false, false);
    }
  }

  // epilogue: bias + swish, C layout: vgpr r holds rows r / r+8
  const int mbase = (lane < 16) ? 0 : 8;
#pragma unroll
  for (int nt = 0; nt < 7; ++nt) {
    int nn = nt*16 + col;
    if (nn < HIDN) {
#pragma unroll
      for (int r = 0; r < 8; ++r) {
        int row = tile*16 + mbase + r;
        if (row < n) {
          float v = acc[nt][r] + lb2[nn];
          h2buf[(size_t)row * HIDN + nn] = v * sigf(v);
        }
      }
    }
  }
}

// ---------------------------------------------------------------------------
// SDE kernel: one thread per sample. mu/t heads, 100 geometric SDE steps,
// then x_hat / mu0 / t0 heads. Weights are wave-uniform -> read via uniform
// global accesses so clang lowers them to SMEM s_load (scalar operands),
// freeing the VALU/ds pipes in the hot loop. No LDS, no barriers.
// ---------------------------------------------------------------------------
__global__ __launch_bounds__(256) void sde_kernel(
    const float* __restrict__ noise,
    const float* __restrict__ h2buf,
    const float* __restrict__ w_mu, const float* __restrict__ b_mu,
    const float* __restrict__ w_t,  const float* __restrict__ b_t,
    const float* __restrict__ w_dec1, const float* __restrict__ b_dec1,
    const float* __restrict__ w_dec2, const float* __restrict__ b_dec2,
    const float* __restrict__ w_mu0a, const float* __restrict__ b_mu0a,
    const float* __restrict__ w_mu0b, const float* __restrict__ b_mu0b,
    const float* __restrict__ w_t0a, const float* __restrict__ b_t0a,
    const float* __restrict__ w_t0b, const float* __restrict__ b_t0b,
    float* __restrict__ out, int n)
{
  const int i = blockIdx.x * 256 + threadIdx.x;
  if (i >= n) return;

  // ---- mu / t heads from h2 ----
  float m0 = b_mu[0], m1 = b_mu[1], ta = b_t[0];
  const float* hrow = h2buf + (size_t)i * HIDN;
#pragma unroll 4
  for (int k = 0; k < HIDN; ++k) {
    float h = hrow[k];
    m0 = fmaf(h, w_mu[2*k],   m0);
    m1 = fmaf(h, w_mu[2*k+1], m1);
    ta = fmaf(h, w_t[k],      ta);
  }
  float tval = sigf(ta);
  out[(size_t)5*n + 2*i]     = m0;
  out[(size_t)5*n + 2*i + 1] = m1;
  out[(size_t)7*n + i]       = tval;

  float z0 = m0, z1 = m1;
  const float dt  = tval * 0.01f;          // t / N_STEPS
  const float sdt = sqrtf(dt);

  const float bd2_0 = b_dec2[0], bd2_1 = b_dec2[1], bd2_2 = b_dec2[2];

  // ---- 100 SDE steps ----
  for (int st = 0; st < 100; ++st) {
    const float* np = noise + ((size_t)st * n + i) * 2;
    if (st + 1 < 100)
      __builtin_prefetch(noise + ((size_t)(st + 1) * n + i) * 2, 0, 1);
    float dW0 = sdt * np[0];
    float dW1 = sdt * np[1];

    float a20 = bd2_0, a21 = bd2_1, a22 = bd2_2;
    float da00=0.f,da01=0.f,da10=0.f,da11=0.f,da20=0.f,da21=0.f;          // da2[m][i]
    float dd0_00=0.f,dd0_01=0.f,dd0_11=0.f;
    float dd1_00=0.f,dd1_01=0.f,dd1_11=0.f;
    float dd2_00=0.f,dd2_01=0.f,dd2_11=0.f;

#pragma unroll 4
    for (int k = 0; k < HIDN; ++k) {
      float w0 = w_dec1[k], w1 = w_dec1[100+k], bb = b_dec1[k];
      float a  = fmaf(z0, w0, fmaf(z1, w1, bb));
      float sg = sigf(a);
      float h  = a * sg;
      float u  = sg * (1.f - sg);
      float hp = fmaf(a, u, sg);                       // swish'
      float hpp = u * fmaf(a, fmaf(-2.f, sg, 1.f), 2.f); // swish''
      float p0 = hp*w0, p1 = hp*w1;
      float r0 = hpp*w0, r1 = hpp*w1;
      float q00 = r0*w0, q01 = r0*w1, q11 = r1*w1;
      float c0 = w_dec2[k*3+0], c1 = w_dec2[k*3+1], c2 = w_dec2[k*3+2];
      a20 = fmaf(c0, h, a20); a21 = fmaf(c1, h, a21); a22 = fmaf(c2, h, a22);
      da00 = fmaf(c0,p0,da00); da01 = fmaf(c0,p1,da01);
      da10 = fmaf(c1,p0,da10); da11 = fmaf(c1,p1,da11);
      da20 = fmaf(c2,p0,da20); da21 = fmaf(c2,p1,da21);
      dd0_00=fmaf(c0,q00,dd0_00); dd0_01=fmaf(c0,q01,dd0_01); dd0_11=fmaf(c0,q11,dd0_11);
      dd1_00=fmaf(c1,q00,dd1_00); dd1_01=fmaf(c1,q01,dd1_01); dd1_11=fmaf(c1,q11,dd1_11);
      dd2_00=fmaf(c2,q00,dd2_00); dd2_01=fmaf(c2,q01,dd2_01); dd2_11=fmaf(c2,q11,dd2_11);
    }

    // outer swish: J[m][i], Hessian H[m][ij] of decoder output m
    float J0_0,J0_1,H0_00,H0_01,H0_11;
    float J1_0,J1_1,H1_00,H1_01,H1_11;
    float J2_0,J2_1,H2_00,H2_01,H2_11;
    {
      float a=a20; float sg=sigf(a); float u=sg*(1.f-sg);
      float sp=fmaf(a,u,sg); float spp=u*fmaf(a,fmaf(-2.f,sg,1.f),2.f);
      J0_0=sp*da00; J0_1=sp*da01;
      H0_00=fmaf(spp*da00,da00,sp*dd0_00);
      H0_01=fmaf(spp*da00,da01,sp*dd0_01);
      H0_11=fmaf(spp*da01,da01,sp*dd0_11);
    }
    {
      float a=a21; float sg=sigf(a); float u=sg*(1.f-sg);
      float sp=fmaf(a,u,sg); float spp=u*fmaf(a,fmaf(-2.f,sg,1.f),2.f);
      J1_0=sp*da10; J1_1=sp*da11;
      H1_00=fmaf(spp*da10,da10,sp*dd1_00);
      H1_01=fmaf(spp*da10,da11,sp*dd1_01);
      H1_11=fmaf(spp*da11,da11,sp*dd1_11);
    }
    {
      float a=a22; float sg=sigf(a); float u=sg*(1.f-sg);
      float sp=fmaf(a,u,sg); float spp=u*fmaf(a,fmaf(-2.f,sg,1.f),2.f);
      J2_0=sp*da20; J2_1=sp*da21;
      H2_00=fmaf(spp*da20,da20,sp*dd2_00);
      H2_01=fmaf(spp*da20,da21,sp*dd2_01);
      H2_11=fmaf(spp*da21,da21,sp*dd2_11);
    }

    float g00 = J0_0*J0_0 + J1_0*J1_0 + J2_0*J2_0;
    float g01 = J0_0*J0_1 + J1_0*J1_1 + J2_0*J2_1;
    float g11 = J0_1*J0_1 + J1_1*J1_1 + J2_1*J2_1;

    // D(ab)_l = d g_ab / d z_l
    float D000 = 2.f*(H0_00*J0_0 + H1_00*J1_0 + H2_00*J2_0);
    float D001 = 2.f*(H0_01*J0_0 + H1_01*J1_0 + H2_01*J2_0);
    float D010 = (H0_00*J0_1 + J0_0*H0_01) + (H1_00*J1_1 + J1_0*H1_01) + (H2_00*J2_1 + J2_0*H2_01);
    float D011 = (H0_01*J0_1 + J0_0*H0_11) + (H1_01*J1_1 + J1_0*H1_11) + (H2_01*J2_1 + J2_0*H2_11);
    float D110 = 2.f*(H0_01*J0_1 + H1_01*J1_1 + H2_01*J2_1);
    float D111 = 2.f*(H0_11*J0_1 + H1_11*J1_1 + H2_11*J2_1);

    float idet = __builtin_amdgcn_rcpf(fmaf(g00, g11, -g01*g01));
    float gi00 =  g11*idet, gi01 = -g01*idet, gi11 = g00*idet;

    // S_m_kl = d_l g_km + d_k g_lm - d_m g_kl
    float S0_00 = D000,              S1_00 = 2.f*D010 - D001;
    float S0_01 = D001,              S1_01 = D110;
    float S0_11 = 2.f*D011 - D110,   S1_11 = D111;

    float G0_00 = 0.5f*(gi00*S0_00 + gi01*S1_00);
    float G0_01 = 0.5f*(gi00*S0_01 + gi01*S1_01);
    float G0_11 = 0.5f*(gi00*S0_11 + gi01*S1_11);
    float G1_00 = 0.5f*(gi01*S0_00 + gi11*S1_00);
    float G1_01 = 0.5f*(gi01*S0_01 + gi11*S1_01);
    float G1_11 = 0.5f*(gi01*S0_11 + gi11*S1_11);

    float det0 = 0.5f*(gi00*G0_00 + 2.f*gi01*G0_01 + gi11*G0_11);
    float det1 = 0.5f*(gi00*G1_00 + 2.f*gi01*G1_01 + gi11*G1_11);

    z0 += det0 + gi00*dW0 + gi01*dW1;
    z1 += det1 + gi01*dW0 + gi11*dW1;
  }

  out[2*(size_t)i]     = z0;
  out[2*(size_t)i + 1] = z1;

  // ---- x_hat = decoder(z) ----
  {
    float a0 = bd2_0, a1 = bd2_1, a2 = bd2_2;
#pragma unroll 4
    for (int k = 0; k < HIDN; ++k) {
      float w0 = w_dec1[k], w1 = w_dec1[100+k], bb = b_dec1[k];
      float t_ = fmaf(z0, w0, fmaf(z1, w1, bb));
      float h  = t_ * sigf(t_);
      a0 = fmaf(w_dec2[k*3+0], h, a0);
      a1 = fmaf(w_dec2[k*3+1], h, a1);
      a2 = fmaf(w_dec2[k*3+2], h, a2);
    }
    out[(size_t)2*n + 3*i + 0] = a0 * sigf(a0);
    out[(size_t)2*n + 3*i + 1] = a1 * sigf(a1);
    out[(size_t)2*n + 3*i + 2] = a2 * sigf(a2);
  }
  // ---- mu0 = swish(z@w_mu0a+b)@w_mu0b + b ----
  {
    float a0 = b_mu0b[0], a1 = b_mu0b[1];
#pragma unroll 4
    for (int k = 0; k < HIDN; ++k) {
      float t_ = fmaf(z0, w_mu0a[k], fmaf(z1, w_mu0a[100+k], b_mu0a[k]));
      float h  = t_ * sigf(t_);
      a0 = fmaf(h, w_mu0b[2*k],   a0);
      a1 = fmaf(h, w_mu0b[2*k+1], a1);
    }
    out[(size_t)8*n + 2*i]     = a0;
    out[(size_t)8*n + 2*i + 1] = a1;
  }
  // ---- t0 = swish(z@w_t0a+b)@w_t0b + b ----
  {
    float a0 = b_t0b[0];
#pragma unroll 4
    for (int k = 0; k < HIDN; ++k) {
      float t_ = fmaf(z0, w_t0a[k], fmaf(z1, w_t0a[100+k], b_t0a[k]));
      float h  = t_ * sigf(t_);
      a0 = fmaf(h, w_t0b[k], a0);
    }
    out[(size_t)10*n + i] = a0;
  }
}

extern "C" void kernel_launch(void* const* d_in, const int* in_sizes, int n_in,
                              void* d_out, int out_size, void* d_ws, size_t ws_size,
                              hipStream_t stream) {
  const float* x      = (const float*)d_in[0];
  const float* noise  = (const float*)d_in[1];
  const float* w_enc1 = (const float*)d_in[2];
  const float* b_enc1 = (const float*)d_in[3];
  const float* w_enc2 = (const float*)d_in[4];
  const float* b_enc2 = (const float*)d_in[5];
  const float* w_mu   = (const float*)d_in[6];
  const float* b_mu   = (const float*)d_in[7];
  const float* w_t    = (const float*)d_in[8];
  const float* b_t    = (const float*)d_in[9];
  const float* w_dec1 = (const float*)d_in[10];
  const float* b_dec1 = (const float*)d_in[11];
  const float* w_dec2 = (const float*)d_in[12];
  const float* b_dec2 = (const float*)d_in[13];
  const float* w_mu0a = (const float*)d_in[14];
  const float* b_mu0a = (const float*)d_in[15];
  const float* w_mu0b = (const float*)d_in[16];
  const float* b_mu0b = (const float*)d_in[17];
  const float* w_t0a  = (const float*)d_in[18];
  const float* b_t0a  = (const float*)d_in[19];
  const float* w_t0b  = (const float*)d_in[20];
  const float* b_t0b  = (const float*)d_in[21];
  float* out = (float*)d_out;
  const int n = in_sizes[0] / 3;

  float* h2buf = (float*)d_ws;   // n*100 floats of scratch

  const int tiles = (n + 15) / 16;
  const int encBlocks = (tiles + 7) / 8;
  enc_wmma_kernel<<<encBlocks, 256, 0, stream>>>(x, w_enc1, b_enc1, w_enc2, b_enc2,
                                                 h2buf, n);

  const int sdeBlocks = (n + 255) / 256;
  sde_kernel<<<sdeBlocks, 256, 0, stream>>>(noise, h2buf,
      w_mu, b_mu, w_t, b_t,
      w_dec1, b_dec1, w_dec2, b_dec2,
      w_mu0a, b_mu0a, w_mu0b, b_mu0b,
      w_t0a, b_t0a, w_t0b, b_t0b,
      out, n);
}